// TopoGAT_7756710936736
// MI455X (gfx1250) — compile-verified
//
#include <hip/hip_runtime.h>
#include <hip/hip_bf16.h>

typedef float v2f __attribute__((ext_vector_type(2)));
typedef float v8f __attribute__((ext_vector_type(8)));

#define NEG_SLOPE 0.2f

// ---------------------------------------------------------------------------
// float atomic max via int punning (init value must be -inf)
// ---------------------------------------------------------------------------
__device__ __forceinline__ void atomicMaxF(float* addr, float v) {
    if (v >= 0.0f) {
        atomicMax((int*)addr, __float_as_int(v));
    } else {
        atomicMin((unsigned int*)addr, __float_as_uint(v));
    }
}

__device__ __forceinline__ void edge_sd(const int* __restrict__ ei, int E, int e,
                                        int& s, int& d) {
    if (e < E) { s = ei[e]; d = ei[E + e]; }
    else       { s = e - E; d = e - E; }
}

// ---------------------------------------------------------------------------
// init: m1/m2 <- -inf (contiguous), s1/out1/s2/out2 <- 0 (contiguous)
// ---------------------------------------------------------------------------
__global__ void init_ws_kernel(float* __restrict__ neginf_base, int n_neg,
                               float* __restrict__ zero_base,  int n_zero) {
    int i = blockIdx.x * blockDim.x + threadIdx.x;
    if (i < n_neg)  neginf_base[i] = -__builtin_inff();
    if (i < n_zero) zero_base[i]   = 0.0f;
}

// ---------------------------------------------------------------------------
// GEMM1: h1[N,64] = concat(x[N,120], topo[N,8]) @ W1[128,64]
// fp32 WMMA 16x16x4. One block = 4 waves = one 16-row tile; wave w does the
// 16-wide column tile starting at w*16. K = 128 -> 32 WMMA per wave.
// Out-of-range rows are CLAMPED for loads (branch-free) and masked on store.
// ---------------------------------------------------------------------------
__global__ __launch_bounds__(128)
void gemm1_wmma_kernel(const float* __restrict__ x, const float* __restrict__ topo,
                       const float* __restrict__ W1, float* __restrict__ h1, int N) {
    const int wave   = threadIdx.x >> 5;
    const int lane   = threadIdx.x & 31;
    const int laneLo = lane & 15;
    const bool hi    = lane >= 16;
    const int r0     = blockIdx.x * 16;
    const int j0     = wave * 16;
    const int rowA   = r0 + laneLo;                       // A-matrix row
    const int rowC   = (rowA < N) ? rowA : (N - 1);       // clamped (no branch)
    const int colB   = j0 + laneLo;                       // B/D column

    v8f acc = {};
#pragma unroll
    for (int k0 = 0; k0 < 128; k0 += 4) {
        // A 16x4 fp32 layout: lanes 0-15 -> K = k0+0 (v0), k0+1 (v1)
        //                     lanes 16-31 -> K = k0+2 (v0), k0+3 (v1)
        const int ka0 = k0 + (hi ? 2 : 0);
        const int ka1 = ka0 + 1;
        // (ka0, ka1) pair never straddles the 120 boundary (120 % 4 == 0),
        // and k0 < 120 folds at compile time per unrolled iteration.
        float2 ap;
        if (k0 < 120) ap = *(const float2*)(x    + (size_t)rowC * 120 + ka0);
        else          ap = *(const float2*)(topo + (size_t)rowC * 8   + (ka0 - 120));
        v2f A; A.x = ap.x; A.y = ap.y;
        // B 4x16 fp32: rows striped across lanes per VGPR (K0/K2 in v0, K1/K3 in v1)
        v2f B; B.x = W1[ka0 * 64 + colB]; B.y = W1[ka1 * 64 + colB];
        acc = __builtin_amdgcn_wmma_f32_16x16x4_f32(false, A, false, B,
                                                    (short)0, acc, false, false);
    }
    // D 16x16 fp32: VGPR c -> row r0+c (lanes 0-15) / r0+c+8 (lanes 16-31)
#pragma unroll
    for (int c = 0; c < 8; ++c) {
        const int row = r0 + c + (hi ? 8 : 0);
        if (row < N) h1[(size_t)row * 64 + colB] = acc[c];
    }
}

// ---------------------------------------------------------------------------
// GEMM2: g[N,40] = h2[N,64] @ W2[64,40], cols padded to 48 (3 tiles of 16).
// one wave per (rowTile, colTile); K = 64 -> 16 WMMA per wave.
// Row and padded column are clamped for loads; store masked.
// ---------------------------------------------------------------------------
__global__ __launch_bounds__(128)
void gemm2_wmma_kernel(const float* __restrict__ h2, const float* __restrict__ W2,
                       float* __restrict__ g, int N, int nTiles) {
    const int t = blockIdx.x * 4 + (threadIdx.x >> 5);
    if (t >= nTiles) return;                 // wave-uniform exit: EXEC stays full
    const int rowTile = t / 3;
    const int colTile = t - rowTile * 3;
    const int lane    = threadIdx.x & 31;
    const int laneLo  = lane & 15;
    const bool hi     = lane >= 16;
    const int r0      = rowTile * 16;
    const int colB    = colTile * 16 + laneLo;
    const int colC    = (colB < 40) ? colB : 39;          // clamped load col
    const int rowA    = r0 + laneLo;
    const int rowC    = (rowA < N) ? rowA : (N - 1);      // clamped load row

    v8f acc = {};
#pragma unroll
    for (int k0 = 0; k0 < 64; k0 += 4) {
        const int ka0 = k0 + (hi ? 2 : 0);
        const int ka1 = ka0 + 1;
        const float2 ap = *(const float2*)(h2 + (size_t)rowC * 64 + ka0);
        v2f A; A.x = ap.x; A.y = ap.y;
        v2f B; B.x = W2[ka0 * 40 + colC]; B.y = W2[ka1 * 40 + colC];
        acc = __builtin_amdgcn_wmma_f32_16x16x4_f32(false, A, false, B,
                                                    (short)0, acc, false, false);
    }
#pragma unroll
    for (int c = 0; c < 8; ++c) {
        const int row = r0 + c + (hi ? 8 : 0);
        if (row < N && colB < 40) g[(size_t)row * 40 + colB] = acc[c];
    }
}

// ---------------------------------------------------------------------------
// attention logits, layer 1: al_src1/al_dst1 [N,8]
// ---------------------------------------------------------------------------
__global__ void attn1_kernel(const float* __restrict__ h1,
                             const float* __restrict__ a_src1,
                             const float* __restrict__ a_dst1,
                             float* __restrict__ al_src1,
                             float* __restrict__ al_dst1, int N) {
    const int i = blockIdx.x * blockDim.x + threadIdx.x;
    if (i >= N * 8) return;
    const int n = i >> 3, hd = i & 7;
    const float4 h0 = *(const float4*)(h1 + (size_t)n * 64 + hd * 8);
    const float4 h4 = *(const float4*)(h1 + (size_t)n * 64 + hd * 8 + 4);
    const float4 as0 = *(const float4*)(a_src1 + hd * 8);
    const float4 as4 = *(const float4*)(a_src1 + hd * 8 + 4);
    const float4 ad0 = *(const float4*)(a_dst1 + hd * 8);
    const float4 ad4 = *(const float4*)(a_dst1 + hd * 8 + 4);
    al_src1[i] = h0.x * as0.x + h0.y * as0.y + h0.z * as0.z + h0.w * as0.w +
                 h4.x * as4.x + h4.y * as4.y + h4.z * as4.z + h4.w * as4.w;
    al_dst1[i] = h0.x * ad0.x + h0.y * ad0.y + h0.z * ad0.z + h0.w * ad0.w +
                 h4.x * ad4.x + h4.y * ad4.y + h4.z * ad4.z + h4.w * ad4.w;
}

// ---------------------------------------------------------------------------
// layer-1 edge passes (8 heads)
// ---------------------------------------------------------------------------
__global__ void edge_max1_kernel(const int* __restrict__ ei, int E, int Etot,
                                 const float* __restrict__ al_src1,
                                 const float* __restrict__ al_dst1,
                                 float* __restrict__ m1) {
    const int e = blockIdx.x * blockDim.x + threadIdx.x;
    if (e >= Etot) return;
    int s, d; edge_sd(ei, E, e, s, d);
#pragma unroll
    for (int hd = 0; hd < 8; ++hd) {
        float v = al_src1[s * 8 + hd] + al_dst1[d * 8 + hd];
        v = (v > 0.0f) ? v : NEG_SLOPE * v;
        atomicMaxF(&m1[d * 8 + hd], v);
    }
}

__global__ void edge_sum1_kernel(const int* __restrict__ ei, int E, int Etot,
                                 const float* __restrict__ al_src1,
                                 const float* __restrict__ al_dst1,
                                 const float* __restrict__ m1,
                                 float* __restrict__ s1) {
    const int e = blockIdx.x * blockDim.x + threadIdx.x;
    if (e >= Etot) return;
    int s, d; edge_sd(ei, E, e, s, d);
#pragma unroll
    for (int hd = 0; hd < 8; ++hd) {
        float v = al_src1[s * 8 + hd] + al_dst1[d * 8 + hd];
        v = (v > 0.0f) ? v : NEG_SLOPE * v;
        atomicAdd(&s1[d * 8 + hd], __expf(v - m1[d * 8 + hd]));
    }
}

__global__ void edge_agg1_kernel(const int* __restrict__ ei, int E, int Etot,
                                 const float* __restrict__ al_src1,
                                 const float* __restrict__ al_dst1,
                                 const float* __restrict__ m1,
                                 const float* __restrict__ s1,
                                 const float* __restrict__ h1,
                                 float* __restrict__ out1) {
    const int i = blockIdx.x * blockDim.x + threadIdx.x;
    if (i >= Etot * 8) return;
    const int e = i >> 3, hd = i & 7;
    int s, d; edge_sd(ei, E, e, s, d);
    float v = al_src1[s * 8 + hd] + al_dst1[d * 8 + hd];
    v = (v > 0.0f) ? v : NEG_SLOPE * v;
    const float alpha = __expf(v - m1[d * 8 + hd]) / s1[d * 8 + hd];
    const float4 h0 = *(const float4*)(h1 + (size_t)s * 64 + hd * 8);
    const float4 h4 = *(const float4*)(h1 + (size_t)s * 64 + hd * 8 + 4);
    float* o = out1 + (size_t)d * 64 + hd * 8;
    atomicAdd(o + 0, alpha * h0.x);
    atomicAdd(o + 1, alpha * h0.y);
    atomicAdd(o + 2, alpha * h0.z);
    atomicAdd(o + 3, alpha * h0.w);
    atomicAdd(o + 4, alpha * h4.x);
    atomicAdd(o + 5, alpha * h4.y);
    atomicAdd(o + 6, alpha * h4.z);
    atomicAdd(o + 7, alpha * h4.w);
}

// ---------------------------------------------------------------------------
// h2 = elu(out1 + b1)
// ---------------------------------------------------------------------------
__global__ void bias_elu_kernel(const float* __restrict__ out1,
                                const float* __restrict__ b1,
                                float* __restrict__ h2, int N) {
    const int i = blockIdx.x * blockDim.x + threadIdx.x;
    if (i >= N * 64) return;
    const float v = out1[i] + b1[i & 63];
    h2[i] = (v > 0.0f) ? v : (__expf(v) - 1.0f);
}

// ---------------------------------------------------------------------------
// attention logits, layer 2 (1 head, 40 features)
// ---------------------------------------------------------------------------
__global__ void attn2_kernel(const float* __restrict__ g,
                             const float* __restrict__ a_src2,
                             const float* __restrict__ a_dst2,
                             float* __restrict__ al_src2,
                             float* __restrict__ al_dst2, int N) {
    const int n = blockIdx.x * blockDim.x + threadIdx.x;
    if (n >= N) return;
    float s = 0.0f, d = 0.0f;
#pragma unroll
    for (int c = 0; c < 40; c += 4) {
        const float4 gv = *(const float4*)(g + (size_t)n * 40 + c);
        const float4 sv = *(const float4*)(a_src2 + c);
        const float4 dv = *(const float4*)(a_dst2 + c);
        s += gv.x * sv.x + gv.y * sv.y + gv.z * sv.z + gv.w * sv.w;
        d += gv.x * dv.x + gv.y * dv.y + gv.z * dv.z + gv.w * dv.w;
    }
    al_src2[n] = s;
    al_dst2[n] = d;
}

__global__ void edge_max2_kernel(const int* __restrict__ ei, int E, int Etot,
                                 const float* __restrict__ al_src2,
                                 const float* __restrict__ al_dst2,
                                 float* __restrict__ m2) {
    const int e = blockIdx.x * blockDim.x + threadIdx.x;
    if (e >= Etot) return;
    int s, d; edge_sd(ei, E, e, s, d);
    float v = al_src2[s] + al_dst2[d];
    v = (v > 0.0f) ? v : NEG_SLOPE * v;
    atomicMaxF(&m2[d], v);
}

__global__ void edge_sum2_kernel(const int* __restrict__ ei, int E, int Etot,
                                 const float* __restrict__ al_src2,
                                 const float* __restrict__ al_dst2,
                                 const float* __restrict__ m2,
                                 float* __restrict__ s2) {
    const int e = blockIdx.x * blockDim.x + threadIdx.x;
    if (e >= Etot) return;
    int s, d; edge_sd(ei, E, e, s, d);
    float v = al_src2[s] + al_dst2[d];
    v = (v > 0.0f) ? v : NEG_SLOPE * v;
    atomicAdd(&s2[d], __expf(v - m2[d]));
}

__global__ void edge_agg2_kernel(const int* __restrict__ ei, int E, int Etot,
                                 const float* __restrict__ al_src2,
                                 const float* __restrict__ al_dst2,
                                 const float* __restrict__ m2,
                                 const float* __restrict__ s2,
                                 const float* __restrict__ g,
                                 float* __restrict__ out2) {
    const int i = blockIdx.x * blockDim.x + threadIdx.x;
    if (i >= Etot * 5) return;
    const int e = i / 5;
    const int d0 = (i - e * 5) * 8;
    int s, d; edge_sd(ei, E, e, s, d);
    float v = al_src2[s] + al_dst2[d];
    v = (v > 0.0f) ? v : NEG_SLOPE * v;
    const float alpha = __expf(v - m2[d]) / s2[d];
    const float4 g0 = *(const float4*)(g + (size_t)s * 40 + d0);
    const float4 g4 = *(const float4*)(g + (size_t)s * 40 + d0 + 4);
    float* o = out2 + (size_t)d * 40 + d0;
    atomicAdd(o + 0, alpha * g0.x);
    atomicAdd(o + 1, alpha * g0.y);
    atomicAdd(o + 2, alpha * g0.z);
    atomicAdd(o + 3, alpha * g0.w);
    atomicAdd(o + 4, alpha * g4.x);
    atomicAdd(o + 5, alpha * g4.y);
    atomicAdd(o + 6, alpha * g4.z);
    atomicAdd(o + 7, alpha * g4.w);
}

// ---------------------------------------------------------------------------
// final: out[n,:] = log_softmax(out2[n,:] + b2)
// ---------------------------------------------------------------------------
__global__ void logsoftmax_kernel(const float* __restrict__ out2,
                                  const float* __restrict__ b2,
                                  float* __restrict__ out, int N) {
    const int n = blockIdx.x * blockDim.x + threadIdx.x;
    if (n >= N) return;
    float v[40];
    float m = -__builtin_inff();
#pragma unroll
    for (int c = 0; c < 40; c += 4) {
        const float4 ov = *(const float4*)(out2 + (size_t)n * 40 + c);
        const float4 bv = *(const float4*)(b2 + c);
        v[c + 0] = ov.x + bv.x; v[c + 1] = ov.y + bv.y;
        v[c + 2] = ov.z + bv.z; v[c + 3] = ov.w + bv.w;
    }
#pragma unroll
    for (int c = 0; c < 40; ++c) m = fmaxf(m, v[c]);
    float sum = 0.0f;
#pragma unroll
    for (int c = 0; c < 40; ++c) sum += __expf(v[c] - m);
    const float lse = m + __logf(sum);
#pragma unroll
    for (int c = 0; c < 40; ++c) out[(size_t)n * 40 + c] = v[c] - lse;
}

// ---------------------------------------------------------------------------
// host launcher
// ---------------------------------------------------------------------------
extern "C" void kernel_launch(void* const* d_in, const int* in_sizes, int n_in,
                              void* d_out, int out_size, void* d_ws, size_t ws_size,
                              hipStream_t stream) {
    const float* x       = (const float*)d_in[0];
    const float* topo    = (const float*)d_in[1];
    const int*   ei      = (const int*)  d_in[2];
    const float* W1      = (const float*)d_in[3];
    const float* a_src1  = (const float*)d_in[4];
    const float* a_dst1  = (const float*)d_in[5];
    const float* b1      = (const float*)d_in[6];
    const float* W2      = (const float*)d_in[7];
    const float* a_src2  = (const float*)d_in[8];
    const float* a_dst2  = (const float*)d_in[9];
    const float* b2      = (const float*)d_in[10];
    float* out = (float*)d_out;

    const int N    = in_sizes[0] / 120;
    const int E    = in_sizes[2] / 2;
    const int Etot = E + N;

    // workspace layout (floats)
    float* ws      = (float*)d_ws;
    float* h1      = ws;                    // N*64
    float* al_src1 = ws + (size_t)N * 64;   // N*8
    float* al_dst1 = ws + (size_t)N * 72;   // N*8
    float* h2      = ws + (size_t)N * 80;   // N*64
    float* g       = ws + (size_t)N * 144;  // N*40
    float* al_src2 = ws + (size_t)N * 184;  // N
    float* al_dst2 = ws + (size_t)N * 185;  // N
    float* m1      = ws + (size_t)N * 186;  // N*8   (-inf init)
    float* m2      = ws + (size_t)N * 194;  // N     (-inf init)
    float* s1      = ws + (size_t)N * 195;  // N*8   (zero init)
    float* out1    = ws + (size_t)N * 203;  // N*64  (zero init)
    float* s2      = ws + (size_t)N * 267;  // N     (zero init)
    float* out2    = ws + (size_t)N * 268;  // N*40  (zero init)

    const int TB = 256;
    const int n_neg  = N * 9;    // m1 + m2 contiguous
    const int n_zero = N * 113;  // s1 + out1 + s2 + out2 contiguous
    const int n_init = (n_neg > n_zero) ? n_neg : n_zero;
    init_ws_kernel<<<(n_init + TB - 1) / TB, TB, 0, stream>>>(m1, n_neg, s1, n_zero);

    // ---- layer 1 ----
    const int rowTiles = (N + 15) / 16;
    gemm1_wmma_kernel<<<rowTiles, 128, 0, stream>>>(x, topo, W1, h1, N);
    attn1_kernel<<<(N * 8 + TB - 1) / TB, TB, 0, stream>>>(h1, a_src1, a_dst1,
                                                           al_src1, al_dst1, N);
    edge_max1_kernel<<<(Etot + TB - 1) / TB, TB, 0, stream>>>(ei, E, Etot,
                                                              al_src1, al_dst1, m1);
    edge_sum1_kernel<<<(Etot + TB - 1) / TB, TB, 0, stream>>>(ei, E, Etot,
                                                              al_src1, al_dst1, m1, s1);
    edge_agg1_kernel<<<((size_t)Etot * 8 + TB - 1) / TB, TB, 0, stream>>>(
        ei, E, Etot, al_src1, al_dst1, m1, s1, h1, out1);
    bias_elu_kernel<<<(N * 64 + TB - 1) / TB, TB, 0, stream>>>(out1, b1, h2, N);

    // ---- layer 2 ----
    const int nTiles2 = rowTiles * 3;
    gemm2_wmma_kernel<<<(nTiles2 + 3) / 4, 128, 0, stream>>>(h2, W2, g, N, nTiles2);
    attn2_kernel<<<(N + TB - 1) / TB, TB, 0, stream>>>(g, a_src2, a_dst2,
                                                       al_src2, al_dst2, N);
    edge_max2_kernel<<<(Etot + TB - 1) / TB, TB, 0, stream>>>(ei, E, Etot,
                                                              al_src2, al_dst2, m2);
    edge_sum2_kernel<<<(Etot + TB - 1) / TB, TB, 0, stream>>>(ei, E, Etot,
                                                              al_src2, al_dst2, m2, s2);
    edge_agg2_kernel<<<((size_t)Etot * 5 + TB - 1) / TB, TB, 0, stream>>>(
        ei, E, Etot, al_src2, al_dst2, m2, s2, g, out2);

    logsoftmax_kernel<<<(N + TB - 1) / TB, TB, 0, stream>>>(out2, b2, out, N);
}